// Model_19378892439845
// MI455X (gfx1250) — compile-verified
//
#include <hip/hip_runtime.h>
#include <math.h>

typedef __attribute__((ext_vector_type(2))) float v2f;
typedef __attribute__((ext_vector_type(8))) float v8f;
typedef unsigned int uint32x4 __attribute__((ext_vector_type(4)));
typedef int int32x8 __attribute__((ext_vector_type(8)));
typedef int int32x4 __attribute__((ext_vector_type(4)));

#define BDIM 256
#define TDIM 512
#define KDIM 256
#define BROWS 16      // batch rows per workgroup
#define PSTR 260      // sP row stride (4-word skew: lanes 0..15 hit distinct banks)

// LDS layout (floats, dynamic shared assumed to start at LDS offset 0):
//   sP    @ 0                : BROWS*PSTR   (exp(alpha - m), padded)
//   sA    @ 4160             : BROWS*KDIM   (current alpha)
//   sM    @ 8256             : BROWS        (row maxes)
//   sLog  @ 8272             : 2*BROWS*KDIM (double-buffered logit tile, TDM dest)
#define SP_OFF 0
#define SA_OFF (BROWS * PSTR)
#define SM_OFF (SA_OFF + BROWS * KDIM)
#define SLOG_OFF (SM_OFF + 16)
#define SMEM_FLOATS (SLOG_OFF + 2 * BROWS * KDIM)

// ---------------------------------------------------------------------------
// TDM: async-load one [16 x 256] f32 logit tile (row stride T*K elems) to LDS.
// D# bitfields per CDNA5 ISA. Wave-uniform args.
// clang-23 toolchain: 6-arg builtin (g0, g1, g2, g3, g_ext, cpol).
// ---------------------------------------------------------------------------
__device__ __forceinline__ void tdm_load_tile(const float* gsrc,
                                              unsigned int lds_byte_off) {
    unsigned long long ga = (unsigned long long)(uintptr_t)gsrc;
    uint32x4 g0;
    g0[0] = 1u;                                   // count=1, user descriptor
    g0[1] = lds_byte_off;                         // lds_addr (bytes)
    g0[2] = (unsigned int)ga;                     // global_addr[31:0]
    g0[3] = (unsigned int)((ga >> 32) & 0x1FFFFFFu) | (2u << 30); // [56:32]|type=2
    int32x8 g1;
    g1[0] = (2 << 16);                            // data_size=2 (4B), wg_mask=0
    g1[1] = (KDIM & 0xFFFF) << 16;                // tensor_dim0 = 256 (lo16)
    g1[2] = (BROWS << 16);                        // tensor_dim0 hi=0 | tensor_dim1=16
    g1[3] = (KDIM << 16);                         // tensor_dim1 hi=0 | tile_dim0=256
    g1[4] = BROWS;                                // tile_dim1=16, tile_dim2=0
    g1[5] = TDIM * KDIM;                          // tensor_dim0_stride lo32 (elems)
    g1[6] = 0;                                    // stride hi | dim1_stride lo
    g1[7] = 0;
    int32x4 z4 = {0, 0, 0, 0};
    int32x8 z8 = {0, 0, 0, 0, 0, 0, 0, 0};
    __builtin_amdgcn_tensor_load_to_lds(g0, g1, z4, z4, z8, 0);
}

// ---------------------------------------------------------------------------
// CRF forward recursion: alpha in LDS, exp(W) fragments pinned in VGPRs,
// logsumexp via FP32 WMMA, logit tiles DMA'd by the Tensor Data Mover.
// grid = B/16 workgroups, block = 512 threads (16 waves).
// ---------------------------------------------------------------------------
__global__ __launch_bounds__(512) void crf_forward_kernel(
    const float* __restrict__ logits,    // [B, T, K]
    const float* __restrict__ W,         // [K, K]
    const int*   __restrict__ lengths,   // [B]
    float*       __restrict__ alpha_sum) // [B] out: logsumexp(alpha_final)
{
    extern __shared__ float smem[];
    float* sP   = smem + SP_OFF;
    float* sA   = smem + SA_OFF;
    float* sM   = smem + SM_OFF;
    float* sLog = smem + SLOG_OFF;

    const int tid  = threadIdx.x;
    const int lane = tid & 31;
    const int wave = tid >> 5;             // 0..15
    const int b0   = blockIdx.x * BROWS;

    const int mrow  = lane & 15;           // M row / N col within fragment
    const int khalf = lane >> 4;           // 0: lanes 0-15, 1: lanes 16-31
    const int n0    = wave * 16;           // this wave's output column tile
    const int crow  = khalf ? 8 : 0;       // C/D: lanes 16-31 hold M = v + 8

    // --- B fragments: exp(W) for this wave's 16 columns, pinned in VGPRs.
    //     Invariant over all 511 timesteps -> zero LDS traffic for B.
    v2f bfrag[64];
#pragma unroll
    for (int kk = 0; kk < 64; ++kk) {
        int k = kk * 4;
        bfrag[kk].x = __expf(W[(k + khalf) * KDIM + n0 + mrow]);
        bfrag[kk].y = __expf(W[(k + 2 + khalf) * KDIM + n0 + mrow]);
    }

    // --- alpha(0) = logits[:, 0, :] ---
    for (int idx = tid; idx < BROWS * KDIM; idx += 512) {
        int r = idx >> 8, c = idx & 255;
        sA[r * KDIM + c] = logits[(size_t)(b0 + r) * TDIM * KDIM + c];
    }
    // lengths for the rows this lane's epilogue writes (same for all waves)
    int lenv[8];
#pragma unroll
    for (int v = 0; v < 8; ++v) lenv[v] = lengths[b0 + v + crow];

    // --- TDM preload of logit tile for t=1 into buffer 1 ---
    if (wave == 0)
        tdm_load_tile(&logits[(size_t)b0 * TDIM * KDIM + (size_t)1 * KDIM],
                      (SLOG_OFF + 1 * BROWS * KDIM) * 4u);
    __syncthreads();

    for (int t = 1; t < TDIM; ++t) {
        // ---- phase 1: wave w computes max + exp-rescale of alpha row w ----
        float vals[8];
        float m = -INFINITY;
#pragma unroll
        for (int j = 0; j < 8; ++j) {
            vals[j] = sA[wave * KDIM + lane + j * 32];
            m = fmaxf(m, vals[j]);
        }
#pragma unroll
        for (int off = 16; off; off >>= 1) m = fmaxf(m, __shfl_xor(m, off, 32));
#pragma unroll
        for (int j = 0; j < 8; ++j)
            sP[wave * PSTR + lane + j * 32] = __expf(vals[j] - m);
        if (lane == 0) sM[wave] = m;

        // ---- TDM: kick off tile t+1, then wait so tile t is resident.
        //      In-order TENSORcnt: wait<=1 after issue => previous op done. ----
        if (wave == 0) {
            if (t + 1 < TDIM) {
                tdm_load_tile(
                    &logits[(size_t)b0 * TDIM * KDIM + (size_t)(t + 1) * KDIM],
                    (unsigned int)((SLOG_OFF + ((t + 1) & 1) * BROWS * KDIM) * 4u));
                __builtin_amdgcn_s_wait_tensorcnt(1);
            } else {
                __builtin_amdgcn_s_wait_tensorcnt(0);
            }
        }
        __syncthreads();

        // ---- phase 2: S = exp(alpha-m) @ exp(W); B from registers ----
        v8f acc = {};
        const float* pa = &sP[mrow * PSTR + 2 * khalf];
#pragma unroll
        for (int kk = 0; kk < 64; ++kk) {
            v2f a;
            a.x = pa[kk * 4];
            a.y = pa[kk * 4 + 1];
            acc = __builtin_amdgcn_wmma_f32_16x16x4_f32(
                false, a, false, bfrag[kk], (short)0, acc, false, false);
        }

        // ---- epilogue: alpha_new = logit(LDS tile) + log(S) + m, masked ----
        const float* tile = &sLog[(t & 1) * BROWS * KDIM];
#pragma unroll
        for (int v = 0; v < 8; ++v) {
            int mr = v + crow;
            float na = tile[mr * KDIM + n0 + mrow] + __logf(acc[v]) + sM[mr];
            if (t < lenv[v]) sA[mr * KDIM + (n0 + mrow)] = na;
        }
        __syncthreads();
    }

    // ---- final: alpha_sum[b] = logsumexp(alpha[b, :]) ----
    {
        float vals[8];
        float m = -INFINITY;
#pragma unroll
        for (int j = 0; j < 8; ++j) {
            vals[j] = sA[wave * KDIM + lane + j * 32];
            m = fmaxf(m, vals[j]);
        }
#pragma unroll
        for (int off = 16; off; off >>= 1) m = fmaxf(m, __shfl_xor(m, off, 32));
        float s = 0.f;
#pragma unroll
        for (int j = 0; j < 8; ++j) s += __expf(vals[j] - m);
#pragma unroll
        for (int off = 16; off; off >>= 1) s += __shfl_xor(s, off, 32);
        if (lane == 0) alpha_sum[b0 + wave] = m + __logf(s);
    }
}

// ---------------------------------------------------------------------------
// Gold path score: sum of gold emissions + gold transitions (one wave / batch)
// ---------------------------------------------------------------------------
__global__ __launch_bounds__(32) void crf_gold_kernel(
    const float* __restrict__ logits,
    const float* __restrict__ W,
    const int*   __restrict__ gold,
    const int*   __restrict__ lengths,
    float*       __restrict__ goldscore)
{
    const int b = blockIdx.x;
    const int lane = threadIdx.x;
    const int len = lengths[b];
    float s = 0.f;
    for (int t = lane; t < TDIM; t += 32) {
        if (t < len) {
            int g = gold[b * TDIM + t];
            s += logits[((size_t)b * TDIM + t) * KDIM + g];
            if (t >= 1) {
                int gp = gold[b * TDIM + t - 1];
                s += W[gp * KDIM + g];
            }
        }
    }
#pragma unroll
    for (int off = 16; off; off >>= 1) s += __shfl_xor(s, off, 32);
    if (lane == 0) goldscore[b] = s;
}

// ---------------------------------------------------------------------------
// Final: out = -mean_b(gold[b] - alpha_sum[b])
// ---------------------------------------------------------------------------
__global__ __launch_bounds__(256) void crf_final_kernel(
    const float* __restrict__ goldscore,
    const float* __restrict__ alpha_sum,
    float*       __restrict__ out)
{
    __shared__ float red[BDIM];
    int i = threadIdx.x;
    red[i] = goldscore[i] - alpha_sum[i];
    __syncthreads();
    for (int off = 128; off; off >>= 1) {
        if (i < off) red[i] += red[i + off];
        __syncthreads();
    }
    if (i == 0) out[0] = -red[0] / (float)BDIM;
}

extern "C" void kernel_launch(void* const* d_in, const int* in_sizes, int n_in,
                              void* d_out, int out_size, void* d_ws, size_t ws_size,
                              hipStream_t stream) {
    const float* logits  = (const float*)d_in[0]; // [B,T,K] f32
    const float* W       = (const float*)d_in[1]; // [K,K]   f32
    const int*   gold    = (const int*)d_in[2];   // [B,T]   i32
    const int*   lengths = (const int*)d_in[3];   // [B]     i32
    float* out = (float*)d_out;

    float* alpha_sum = (float*)d_ws;              // B floats
    float* goldscore = alpha_sum + BDIM;          // B floats

    const size_t shmem = (size_t)SMEM_FLOATS * sizeof(float);   // ~64.3 KB
    (void)hipFuncSetAttribute((const void*)crf_forward_kernel,
                              hipFuncAttributeMaxDynamicSharedMemorySize,
                              (int)shmem);

    crf_forward_kernel<<<BDIM / BROWS, 512, shmem, stream>>>(logits, W, lengths,
                                                             alpha_sum);
    crf_gold_kernel<<<BDIM, 32, 0, stream>>>(logits, W, gold, lengths, goldscore);
    crf_final_kernel<<<1, 256, 0, stream>>>(goldscore, alpha_sum, out);
}